// CudaRenderer_18519898980597
// MI455X (gfx1250) — compile-verified
//
#include <hip/hip_runtime.h>
#include <hip/hip_bf16.h>

typedef float v4f __attribute__((ext_vector_type(4)));
typedef int   v4i __attribute__((ext_vector_type(4)));

#define BZc   8
#define Dc    16
#define Hc    512
#define Wc    512
#define HWc   (Hc * Wc)          // 262144
#define NPIXc (BZc * HWc)        // 2097152 pixels
#define NGRP  (NPIXc / 4)        // 524288 groups of 4 pixels
#define BLOCK 256
#define GRID  1024               // stride = 262144 -> exactly 2 groups per thread

// Stage one 4-pixel group's streaming data (48B bary + 16B tri) into LDS via
// CDNA5 async global->LDS copies (ASYNCcnt, in-order completion). NT temporal
// hint: bary/tri are read exactly once, keep them from evicting the
// L2-resident attrs table that the random gathers rely on. The 24-bit inst
// offset is added to BOTH the LDS and the global address.
__device__ __forceinline__ void issue_group(const float* bary, const int* tri, int grp,
                                            unsigned ldsB, unsigned ldsT) {
    unsigned long long ab = (unsigned long long)(const void*)(bary + (size_t)grp * 12u);
    unsigned long long at = (unsigned long long)(const void*)(tri  + (size_t)grp * 4u);
    asm volatile(
        "global_load_async_to_lds_b128 %0, %2, off th:TH_LOAD_NT\n\t"
        "global_load_async_to_lds_b128 %0, %2, off offset:16 th:TH_LOAD_NT\n\t"
        "global_load_async_to_lds_b128 %0, %2, off offset:32 th:TH_LOAD_NT\n\t"
        "global_load_async_to_lds_b128 %1, %3, off th:TH_LOAD_NT"
        :
        : "v"(ldsB), "v"(ldsT), "v"(ab), "v"(at)
        : "memory");
}

__global__ __launch_bounds__(BLOCK) void render_interp_kernel(
    const float* __restrict__ attrs,   // [BZ*NF, 3, 16] f32 (L2 resident, RT)
    const float* __restrict__ bary,    // [NPIX, 3] f32 (streamed, async->LDS, NT)
    const int*   __restrict__ tri,     // [NPIX] i32   (streamed, async->LDS, NT)
    float*       __restrict__ out)     // [BZ, 17, H, W] f32 (NT stores)
{
    __shared__ v4f sBary[2][BLOCK * 3];   // 2 x 12 KB
    __shared__ v4i sTri [2][BLOCK];       // 2 x  4 KB

    const int tid    = threadIdx.x;
    const int stride = GRID * BLOCK;

    // Low 32 bits of a flat pointer into LDS == wave-relative LDS byte address
    // (ISA 10.2: LDS address is the low 32 bits); exactly what the async op's VDST holds.
    unsigned ldsB0 = (unsigned)(size_t)&sBary[0][tid * 3];
    unsigned ldsB1 = (unsigned)(size_t)&sBary[1][tid * 3];
    unsigned ldsT0 = (unsigned)(size_t)&sTri[0][tid];
    unsigned ldsT1 = (unsigned)(size_t)&sTri[1][tid];

    int g   = blockIdx.x * BLOCK + tid;
    int buf = 0;
    if (g < NGRP) issue_group(bary, tri, g, ldsB0, ldsT0);

    for (; g < NGRP; g += stride, buf ^= 1) {
        const int gn = g + stride;
        if (gn < NGRP) {
            // Kick next tile, then wait only for the older 4 async ops
            // (async loads complete in order) -> copy/compute overlap.
            issue_group(bary, tri, gn, buf ? ldsB0 : ldsB1, buf ? ldsT0 : ldsT1);
            asm volatile("s_wait_asynccnt 4" ::: "memory");
        } else {
            asm volatile("s_wait_asynccnt 0" ::: "memory");
        }

        const v4i t4 = sTri[buf][tid];
        const v4f B0 = sBary[buf][tid * 3 + 0];
        const v4f B1 = sBary[buf][tid * 3 + 1];
        const v4f B2 = sBary[buf][tid * 3 + 2];

        const float w[4][3] = {{B0.x, B0.y, B0.z},
                               {B0.w, B1.x, B1.y},
                               {B1.z, B1.w, B2.x},
                               {B2.y, B2.z, B2.w}};

        const v4f zero = {0.0f, 0.0f, 0.0f, 0.0f};
        v4f pd[16];                    // plane-major: pd[d][j] = pixel j, channel d
        #pragma unroll
        for (int q = 0; q < 16; ++q) pd[q] = zero;
        v4f vis = zero;

        const v4f* __restrict__ A4 = (const v4f*)attrs;
        #pragma unroll
        for (int j = 0; j < 4; ++j) {
            const int   t     = t4[j];
            const bool  valid = (t >= 0);
            const int   id    = valid ? t : 0;      // clamp bg to 0 like reference
            const float m     = valid ? 1.0f : 0.0f;
            const float w0 = w[j][0] * m, w1 = w[j][1] * m, w2 = w[j][2] * m;

            // 3 vertices x 16 f32 = 12 x b128 gathers; random but L2-resident.
            const v4f* __restrict__ Ap = A4 + (size_t)id * 12u;
            #pragma unroll
            for (int q = 0; q < 4; ++q) {
                const v4f a = Ap[q], b = Ap[4 + q], c = Ap[8 + q];
                #pragma unroll
                for (int r = 0; r < 4; ++r)
                    pd[q * 4 + r][j] = fmaf(w0, a[r], fmaf(w1, b[r], w2 * c[r]));
            }
            vis[j] = m;
        }

        // Coalesced, non-temporal b128 store per channel plane.
        const int n      = g / (HWc / 4);
        const int hwBase = (g - n * (HWc / 4)) * 4;
        float* ob = out + (size_t)n * 17 * HWc + hwBase;
        #pragma unroll
        for (int d = 0; d < 16; ++d)
            __builtin_nontemporal_store(pd[d], (v4f*)(ob + (size_t)d * HWc));
        __builtin_nontemporal_store(vis, (v4f*)(ob + (size_t)16 * HWc));
    }
}

extern "C" void kernel_launch(void* const* d_in, const int* in_sizes, int n_in,
                              void* d_out, int out_size, void* d_ws, size_t ws_size,
                              hipStream_t stream) {
    (void)in_sizes; (void)n_in; (void)out_size; (void)d_ws; (void)ws_size;
    const float* attrs = (const float*)d_in[0];
    const float* bary  = (const float*)d_in[1];
    const int*   tri   = (const int*)d_in[2];
    float*       out   = (float*)d_out;
    render_interp_kernel<<<dim3(GRID), dim3(BLOCK), 0, stream>>>(attrs, bary, tri, out);
}